// DirectedGraphConv_73358041415911
// MI455X (gfx1250) — compile-verified
//
#include <hip/hip_runtime.h>

typedef __attribute__((ext_vector_type(2))) float v2f;
typedef __attribute__((ext_vector_type(8))) float v8f;

#define Bb 32
#define Nn 128
#define Dn 512
#define NL 16
#define ROWS (Bb * Nn)   // 4096

// ---------------------------------------------------------------------------
// Kernel 1: W_eff[o][i] = W0[o][i] + W1[o][i] + (o==i);  Bt[o][l] = bias[l][o]
// ---------------------------------------------------------------------------
__global__ void prep_weights(const float* __restrict__ w,
                             const float* __restrict__ bias,
                             float* __restrict__ Wa,
                             float* __restrict__ Bt) {
    int idx = blockIdx.x * blockDim.x + threadIdx.x;
    if (idx < Dn * Dn) {
        int o = idx / Dn;
        int i = idx - o * Dn;
        float v = w[idx] + w[Dn * Dn + idx];
        if (o == i) v += 1.0f;
        Wa[idx] = v;
    }
    if (idx < Dn * NL) {
        int o = idx / NL;
        int l = idx - o * NL;
        Bt[idx] = bias[l * Dn + o];
    }
}

// ---------------------------------------------------------------------------
// Kernel 2: per-row 16-bin histogram of edge labels, stored as float.
// One block (128 threads) per (b,n) row; LDS atomics.
// ---------------------------------------------------------------------------
__global__ void count_labels(const int* __restrict__ graph,
                             float* __restrict__ cnt) {
    __shared__ int h[NL];
    int row = blockIdx.x;          // 0..4095
    int t = threadIdx.x;           // 0..127
    if (t < NL) h[t] = 0;
    __syncthreads();
    int lab = graph[row * Nn + t] & (NL - 1);
    atomicAdd(&h[lab], 1);
    __syncthreads();
    if (t < NL) cnt[row * NL + t] = (float)h[t];
}

// ---------------------------------------------------------------------------
// Kernel 3: fused WMMA GEMM.
//   out[m][o] = sum_k X[m][k]*Wa[o][k]  (K=512, includes identity via diag)
//             + sum_l cnt[m][l]*Bt[o][l] (K=16 tail = edge-label bias)
// Each wave: 32(M) x 64(N) region = 2x4 tiles of 16x16, f32 WMMA 16x16x4.
// Block = 256 threads = 8 waves; grid = 128 blocks -> 1024 waves total.
// ---------------------------------------------------------------------------
__global__ __launch_bounds__(256) void gemm_fused(
    const float* __restrict__ X,
    const float* __restrict__ Wa,
    const float* __restrict__ Bt,
    const float* __restrict__ cnt,
    float* __restrict__ out) {

    const int lane = threadIdx.x & 31;
    const int wave = threadIdx.x >> 5;
    const int wid  = blockIdx.x * 8 + wave;      // 0..1023
    const int m0 = (wid >> 3) * 32;              // 128 M-blocks of 32
    const int o0 = (wid & 7) * 64;               // 8 N-blocks of 64
    const int r = lane & 15;
    const int h = lane >> 4;

    // Base pointers for A fragments (two M-tiles) and B fragments (four N-tiles)
    const float* a0p = X + (size_t)(m0 + r) * Dn + 2 * h;
    const float* a1p = X + (size_t)(m0 + 16 + r) * Dn + 2 * h;
    const float* bp0 = Wa + (size_t)(o0 + 0 * 16 + r) * Dn + 2 * h;
    const float* bp1 = Wa + (size_t)(o0 + 1 * 16 + r) * Dn + 2 * h;
    const float* bp2 = Wa + (size_t)(o0 + 2 * 16 + r) * Dn + 2 * h;
    const float* bp3 = Wa + (size_t)(o0 + 3 * 16 + r) * Dn + 2 * h;

    v8f acc[2][4];
#pragma unroll
    for (int mt = 0; mt < 2; ++mt)
#pragma unroll
        for (int nt = 0; nt < 4; ++nt)
            acc[mt][nt] = (v8f){0.f, 0.f, 0.f, 0.f, 0.f, 0.f, 0.f, 0.f};

    // Main K loop over the 512-wide feature dimension, 4 per step.
#pragma unroll 4
    for (int k0 = 0; k0 < Dn; k0 += 4) {
        v2f a0 = *(const v2f*)(a0p + k0);
        v2f a1 = *(const v2f*)(a1p + k0);
        v2f b0 = *(const v2f*)(bp0 + k0);
        v2f b1 = *(const v2f*)(bp1 + k0);
        v2f b2 = *(const v2f*)(bp2 + k0);
        v2f b3 = *(const v2f*)(bp3 + k0);
        acc[0][0] = __builtin_amdgcn_wmma_f32_16x16x4_f32(false, a0, false, b0, (short)0, acc[0][0], false, false);
        acc[1][0] = __builtin_amdgcn_wmma_f32_16x16x4_f32(false, a1, false, b0, (short)0, acc[1][0], false, false);
        acc[0][1] = __builtin_amdgcn_wmma_f32_16x16x4_f32(false, a0, false, b1, (short)0, acc[0][1], false, false);
        acc[1][1] = __builtin_amdgcn_wmma_f32_16x16x4_f32(false, a1, false, b1, (short)0, acc[1][1], false, false);
        acc[0][2] = __builtin_amdgcn_wmma_f32_16x16x4_f32(false, a0, false, b2, (short)0, acc[0][2], false, false);
        acc[1][2] = __builtin_amdgcn_wmma_f32_16x16x4_f32(false, a1, false, b2, (short)0, acc[1][2], false, false);
        acc[0][3] = __builtin_amdgcn_wmma_f32_16x16x4_f32(false, a0, false, b3, (short)0, acc[0][3], false, false);
        acc[1][3] = __builtin_amdgcn_wmma_f32_16x16x4_f32(false, a1, false, b3, (short)0, acc[1][3], false, false);
    }

    // Bias tail: K = 16 over (counts x transposed-bias); leading dim = 16.
#pragma unroll
    for (int k0 = 0; k0 < NL; k0 += 4) {
        v2f a0 = *(const v2f*)(cnt + (size_t)(m0 + r) * NL + k0 + 2 * h);
        v2f a1 = *(const v2f*)(cnt + (size_t)(m0 + 16 + r) * NL + k0 + 2 * h);
#pragma unroll
        for (int nt = 0; nt < 4; ++nt) {
            v2f b = *(const v2f*)(Bt + (size_t)(o0 + nt * 16 + r) * NL + k0 + 2 * h);
            acc[0][nt] = __builtin_amdgcn_wmma_f32_16x16x4_f32(false, a0, false, b, (short)0, acc[0][nt], false, false);
            acc[1][nt] = __builtin_amdgcn_wmma_f32_16x16x4_f32(false, a1, false, b, (short)0, acc[1][nt], false, false);
        }
    }

    // Store: acc VGPR v, half h -> (M = m0 + mt*16 + v + 8h, O = o0 + nt*16 + r)
#pragma unroll
    for (int mt = 0; mt < 2; ++mt)
#pragma unroll
        for (int nt = 0; nt < 4; ++nt)
#pragma unroll
            for (int v = 0; v < 8; ++v)
                out[(size_t)(m0 + mt * 16 + v + 8 * h) * Dn + o0 + nt * 16 + r] = acc[mt][nt][v];
}

// ---------------------------------------------------------------------------
extern "C" void kernel_launch(void* const* d_in, const int* in_sizes, int n_in,
                              void* d_out, int out_size, void* d_ws, size_t ws_size,
                              hipStream_t stream) {
    const float* feature = (const float*)d_in[0];   // [32,128,512]
    const int*   graph   = (const int*)d_in[1];     // [32,128,128] labels 0..15
    const float* weights = (const float*)d_in[2];   // [2,512,512]
    const float* bias    = (const float*)d_in[3];   // [16,512]
    float* out = (float*)d_out;                     // [32,128,512]

    float* ws = (float*)d_ws;
    float* Wa  = ws;                                // 512*512   = 262144 floats
    float* Bt  = ws + Dn * Dn;                      // 512*16    =   8192 floats
    float* cnt = ws + Dn * Dn + Dn * NL;            // 4096*16   =  65536 floats
                                                    // total ~1.31 MB of d_ws

    prep_weights<<<(Dn * Dn + 255) / 256, 256, 0, stream>>>(weights, bias, Wa, Bt);
    count_labels<<<ROWS, Nn, 0, stream>>>(graph, cnt);
    gemm_fused<<<128, 256, 0, stream>>>(feature, Wa, Bt, cnt, out);
}